// ComponentROIModel_90348932039057
// MI455X (gfx1250) — compile-verified
//
#include <hip/hip_runtime.h>
#include <hip/hip_bf16.h>

typedef __attribute__((ext_vector_type(2))) float v2f;
typedef __attribute__((ext_vector_type(8))) float v8f;

#define IMG 512
#define BIMG 8
#define CIN 384
#define CF 256
#define HF 16
#define WF 16
#define HWF (HF*WF)       // 256
#define PP 7
#define NBOXN 1024
#define KRF (CF*PP*PP)    // 12544
#define CMID 128
#define CAT_K (3*KRF)     // 37632
#define OVN 512
#define ROI_SCALE (16.0f/512.0f)

// qhead GEMM tiling
#define KC  32            // K chunk staged in LDS
#define AST 36            // LDS A row stride (floats): 16B aligned, conflict-free
#define BST 72            // LDS B row stride (floats): 16B aligned, conflict-free

// ---------------------------------------------------------------------------
// 1) FPN 1x1 conv + ReLU as WMMA GEMM: M=2048 pixels, N=256, K=384
//    A[p][k] = feat[b][k][hw]  (p = b*256 + hw),  B[k][n] = fpn_w[n][k]
// ---------------------------------------------------------------------------
__global__ void fpn_conv_wmma(const float* __restrict__ feat,
                              const float* __restrict__ w,     // [256,384]
                              const float* __restrict__ bias,  // [256]
                              float* __restrict__ fpn) {       // [8,256,16,16]
  const int lane  = threadIdx.x;            // 0..31
  const int n0    = blockIdx.x * 16;        // 16 N-tiles
  const int m0    = blockIdx.y * 16;        // 128 M-tiles
  const int lrow  = lane & 15;
  const int khalf = (lane >> 4) << 1;       // 0 or 2 (K split per half-wave)
  const int row   = m0 + lrow;
  const int b     = row >> 8;
  const int hw    = row & 255;
  const int col   = n0 + lrow;
  const float* fbase = feat + (size_t)b * CIN * HWF + hw;
  const float* wbase = w + (size_t)col * CIN;
  v8f acc = {};
  for (int k = 0; k < CIN; k += 4) {
    const int ka = k + khalf;
    v2f a, bb;
    a.x  = fbase[(size_t)ka * HWF];
    a.y  = fbase[(size_t)(ka + 1) * HWF];
    bb.x = wbase[ka];
    bb.y = wbase[ka + 1];
    acc = __builtin_amdgcn_wmma_f32_16x16x4_f32(false, a, false, bb,
                                                (short)0, acc, false, false);
  }
  const float bv = bias[col];
  const int   mh = (lane >> 4) << 3;        // rows j (lanes<16) / j+8 (lanes>=16)
#pragma unroll
  for (int j = 0; j < 8; ++j) {
    const int m   = m0 + j + mh;
    const int ob  = m >> 8, ohw = m & 255;
    const float v = acc[j] + bv;
    fpn[(size_t)ob * CF * HWF + (size_t)col * HWF + ohw] = v > 0.f ? v : 0.f;
  }
}

// ---------------------------------------------------------------------------
// 2) 3x3 conv 256->128 + ReLU (SAME). Block = one (b, co), 256 px per block.
// ---------------------------------------------------------------------------
__global__ void seg_conv3x3(const float* __restrict__ fpn,
                            const float* __restrict__ w,     // [128,256,3,3]
                            const float* __restrict__ bias,  // [128]
                            float* __restrict__ mid) {       // [8,128,16,16]
  const int co  = blockIdx.x & (CMID - 1);
  const int b   = blockIdx.x >> 7;
  const int tid = threadIdx.x;
  const int h   = tid >> 4, x = tid & 15;
  const float* fb = fpn + (size_t)b * CF * HWF;
  const float* wb = w + (size_t)co * CF * 9;
  float acc = bias[co];
  for (int ci = 0; ci < CF; ++ci) {
    const float* fc = fb + (size_t)ci * HWF;
    const float* wc = wb + ci * 9;
#pragma unroll
    for (int ky = 0; ky < 3; ++ky) {
      const int yy = h + ky - 1;
      if (yy < 0 || yy >= HF) continue;
#pragma unroll
      for (int kx = 0; kx < 3; ++kx) {
        const int xx = x + kx - 1;
        if (xx < 0 || xx >= WF) continue;
        acc += fc[yy * WF + xx] * wc[ky * 3 + kx];
      }
    }
  }
  mid[(size_t)b * CMID * HWF + (size_t)co * HWF + tid] = acc > 0.f ? acc : 0.f;
}

// ---------------------------------------------------------------------------
// 3) 1x1 conv 128->4 (no ReLU). 8192 outputs.
// ---------------------------------------------------------------------------
__global__ void seg_conv1x1(const float* __restrict__ mid,
                            const float* __restrict__ w,     // [4,128]
                            const float* __restrict__ bias,  // [4]
                            float* __restrict__ seg) {       // [8,4,16,16]
  const int o  = blockIdx.x * blockDim.x + threadIdx.x;
  const int hw = o & 255;
  const int co = (o >> 8) & 3;
  const int b  = o >> 10;
  const float* mb = mid + (size_t)b * CMID * HWF + hw;
  const float* wb = w + co * CMID;
  float acc = bias[co];
  for (int ci = 0; ci < CMID; ++ci) acc += mb[(size_t)ci * HWF] * wb[ci];
  seg[o] = acc;
}

// ---------------------------------------------------------------------------
// 4) Bilinear upsample 16x16 -> 512x512 (half-pixel, clamped edges).
// ---------------------------------------------------------------------------
__global__ void seg_upsample(const float* __restrict__ seg,   // [8,4,16,16]
                             float* __restrict__ out) {       // [8,4,512,512]
  const size_t o = (size_t)blockIdx.x * blockDim.x + threadIdx.x;
  const int    ox = (int)(o & 511);
  const int    oy = (int)((o >> 9) & 511);
  const size_t pl = o >> 18;                 // b*4 + c
  const float sy = (oy + 0.5f) * (16.0f / 512.0f) - 0.5f;
  const float sx = (ox + 0.5f) * (16.0f / 512.0f) - 0.5f;
  const float fy = floorf(sy), fx = floorf(sx);
  const float ly = sy - fy,    lx = sx - fx;
  const int y0 = (int)fy, x0 = (int)fx;
  const int y0c = min(max(y0, 0), 15), y1c = min(max(y0 + 1, 0), 15);
  const int x0c = min(max(x0, 0), 15), x1c = min(max(x0 + 1, 0), 15);
  const float* p = seg + pl * HWF;
  const float v = (1.f - ly) * ((1.f - lx) * p[y0c * 16 + x0c] + lx * p[y0c * 16 + x1c])
                +        ly  * ((1.f - lx) * p[y1c * 16 + x0c] + lx * p[y1c * 16 + x1c]);
  out[o] = v;
}

// ---------------------------------------------------------------------------
// 5) roi_align: block = box, thread = channel. rf[n][c*49 + ph*7 + pw]
// ---------------------------------------------------------------------------
__global__ void roi_align_kernel(const float* __restrict__ fpn,
                                 const float* __restrict__ boxes,  // [1024,5]
                                 float* __restrict__ rf) {         // [1024,12544]
  const int n = blockIdx.x;
  const int c = threadIdx.x;                // 0..255
  const float* bx = boxes + n * 5;
  const int   bi = (int)bx[0];
  const float x1 = bx[1] * ROI_SCALE, y1 = bx[2] * ROI_SCALE;
  const float x2 = bx[3] * ROI_SCALE, y2 = bx[4] * ROI_SCALE;
  const float rw = fmaxf(x2 - x1, 1.0f), rh = fmaxf(y2 - y1, 1.0f);
  const float bw = rw * (1.0f / PP), bh = rh * (1.0f / PP);
  const float* f = fpn + (size_t)bi * CF * HWF + (size_t)c * HWF;
  float* r = rf + (size_t)n * KRF + (size_t)c * (PP * PP);
  for (int ph = 0; ph < PP; ++ph) {
    for (int pw = 0; pw < PP; ++pw) {
      float acc = 0.f;
#pragma unroll
      for (int iy = 0; iy < 2; ++iy) {
        const float yc  = y1 + (ph + 0.25f + 0.5f * iy) * bh;
        const bool  vy  = (yc > -1.0f) && (yc < 16.0f);
        const float ycl = fminf(fmaxf(yc, 0.f), 15.f);
        const int   yy0 = (int)floorf(ycl);
        const int   yy1 = min(yy0 + 1, 15);
        const float ly  = ycl - (float)yy0;
#pragma unroll
        for (int ix = 0; ix < 2; ++ix) {
          const float xc  = x1 + (pw + 0.25f + 0.5f * ix) * bw;
          const bool  vx  = (xc > -1.0f) && (xc < 16.0f);
          const float xcl = fminf(fmaxf(xc, 0.f), 15.f);
          const int   xx0 = (int)floorf(xcl);
          const int   xx1 = min(xx0 + 1, 15);
          const float lx  = xcl - (float)xx0;
          const float v00 = f[yy0 * 16 + xx0], v01 = f[yy0 * 16 + xx1];
          const float v10 = f[yy1 * 16 + xx0], v11 = f[yy1 * 16 + xx1];
          const float v = (1.f - ly) * ((1.f - lx) * v00 + lx * v01)
                        +        ly  * ((1.f - lx) * v10 + lx * v11);
          acc += (vy && vx) ? v : 0.f;
        }
      }
      r[ph * PP + pw] = acc * 0.25f;
    }
  }
}

// ---------------------------------------------------------------------------
// 6) qhead GEMM: hidden = ReLU(rf[1024,12544] @ w1[12544,256] + b1)
//    LDS-tiled: block = 8 waves = 128(M) x 64(N) tile, K chunked by 32.
//    Cooperative coalesced float4 global loads -> LDS (b128 ds stores),
//    WMMA fragments read from LDS with conflict-free padded strides.
// ---------------------------------------------------------------------------
__global__ __launch_bounds__(256)
void qhead_gemm_wmma(const float* __restrict__ rf,
                     const float* __restrict__ w1,   // [12544,256]
                     const float* __restrict__ b1,   // [256]
                     float* __restrict__ hidden) {   // [1024,256]
  __shared__ float ldsA[128 * AST];   // 18432 B
  __shared__ float ldsB[KC * BST];    //  9216 B
  const int tid   = threadIdx.x;
  const int lane  = tid & 31;
  const int wv    = tid >> 5;          // wave 0..7 -> M strip
  const int n0    = blockIdx.x * 64;
  const int mB    = blockIdx.y * 128;
  const int lrow  = lane & 15;
  const int khalf = (lane >> 4) << 1;  // 0 or 2

  // cooperative staging assignments
  const int ar = tid >> 1;             // 0..127  A row
  const int ac = (tid & 1) * 16;       // 0/16    A col offset (16 floats each)
  const int br = tid >> 3;             // 0..31   B row (k)
  const int bc = (tid & 7) * 8;        // 0..56   B col offset (8 floats each)
  const float* agp = rf + (size_t)(mB + ar) * KRF + ac;
  const float* bgp = w1 + (size_t)br * CF + n0 + bc;
  float* asp = ldsA + ar * AST + ac;
  float* bsp = ldsB + br * BST + bc;

  const float* afrag = ldsA + (wv * 16 + lrow) * AST;
  const float* bfrag = ldsB + lrow;

  v8f acc0 = {}, acc1 = {}, acc2 = {}, acc3 = {};
  for (int k0 = 0; k0 < KRF; k0 += KC) {
    __syncthreads();                    // previous chunk fully consumed
    const float4 a0 = *(const float4*)(agp + k0 + 0);
    const float4 a1 = *(const float4*)(agp + k0 + 4);
    const float4 a2 = *(const float4*)(agp + k0 + 8);
    const float4 a3 = *(const float4*)(agp + k0 + 12);
    const float4 g0 = *(const float4*)(bgp + (size_t)k0 * CF + 0);
    const float4 g1 = *(const float4*)(bgp + (size_t)k0 * CF + 4);
    *(float4*)(asp + 0)  = a0;
    *(float4*)(asp + 4)  = a1;
    *(float4*)(asp + 8)  = a2;
    *(float4*)(asp + 12) = a3;
    *(float4*)(bsp + 0)  = g0;
    *(float4*)(bsp + 4)  = g1;
    __syncthreads();                    // chunk visible to all waves
#pragma unroll
    for (int kk = 0; kk < KC; kk += 4) {
      const int ka = kk + khalf;
      const v2f a = *(const v2f*)(afrag + ka);
      v2f bb;
      bb.x = bfrag[ka * BST + 0];       bb.y = bfrag[(ka + 1) * BST + 0];
      acc0 = __builtin_amdgcn_wmma_f32_16x16x4_f32(false, a, false, bb, (short)0, acc0, false, false);
      bb.x = bfrag[ka * BST + 16];      bb.y = bfrag[(ka + 1) * BST + 16];
      acc1 = __builtin_amdgcn_wmma_f32_16x16x4_f32(false, a, false, bb, (short)0, acc1, false, false);
      bb.x = bfrag[ka * BST + 32];      bb.y = bfrag[(ka + 1) * BST + 32];
      acc2 = __builtin_amdgcn_wmma_f32_16x16x4_f32(false, a, false, bb, (short)0, acc2, false, false);
      bb.x = bfrag[ka * BST + 48];      bb.y = bfrag[(ka + 1) * BST + 48];
      acc3 = __builtin_amdgcn_wmma_f32_16x16x4_f32(false, a, false, bb, (short)0, acc3, false, false);
    }
  }
  const int mh = (lane >> 4) << 3;
  const int mw = mB + wv * 16;
#pragma unroll
  for (int t = 0; t < 4; ++t) {
    const v8f acc = (t == 0) ? acc0 : (t == 1) ? acc1 : (t == 2) ? acc2 : acc3;
    const int n = n0 + t * 16 + lrow;
    const float bv = b1[n];
#pragma unroll
    for (int j = 0; j < 8; ++j) {
      const int m = mw + j + mh;
      const float v = acc[j] + bv;
      hidden[(size_t)m * CF + n] = v > 0.f ? v : 0.f;
    }
  }
}

// ---------------------------------------------------------------------------
// 7) qual = hidden @ w2 + b2   (one thread per box)
// ---------------------------------------------------------------------------
__global__ void qual_kernel(const float* __restrict__ hidden,
                            const float* __restrict__ w2,  // [256]
                            const float* __restrict__ b2,  // [1]
                            float* __restrict__ qout) {    // [1024]
  const int n = blockIdx.x * blockDim.x + threadIdx.x;
  const float* h = hidden + (size_t)n * CF;
  float acc = b2[0];
  for (int j = 0; j < CF; ++j) acc += h[j] * w2[j];
  qout[n] = acc;
}

// ---------------------------------------------------------------------------
// 8) per-image mean of rf columns -> cat[16, 37632] (rows 8..15 zero-padded
//    so the overall WMMA kernel can run with EXEC all-1s, M padded to 16).
//    Deterministic segmented sum (no float atomics -> bitwise-stable replays).
// ---------------------------------------------------------------------------
__global__ void seg_avg_kernel(const float* __restrict__ rf,
                               const float* __restrict__ boxes,
                               float* __restrict__ cat, int head) {
  const int k = blockIdx.x * blockDim.x + threadIdx.x;  // 0..12543
  float acc[8], cnt[8];
#pragma unroll
  for (int i = 0; i < 8; ++i) { acc[i] = 0.f; cnt[i] = 0.f; }
  for (int n = 0; n < NBOXN; ++n) {
    const int   im = (int)boxes[n * 5];        // scalar (uniform) load
    const float v  = rf[(size_t)n * KRF + k];
#pragma unroll
    for (int i = 0; i < 8; ++i)
      if (im == i) { acc[i] += v; cnt[i] += 1.f; }
  }
#pragma unroll
  for (int i = 0; i < 8; ++i)
    cat[(size_t)i * CAT_K + head * KRF + k] = acc[i] / fmaxf(cnt[i], 1.f);
#pragma unroll
  for (int i = 8; i < 16; ++i)
    cat[(size_t)i * CAT_K + head * KRF + k] = 0.f;   // zero pad rows 8..15
}

// ---------------------------------------------------------------------------
// 9) overall GEMM: hidden2 = ReLU(cat[16,37632] @ ov_w1[37632,512] + ov_b1)
//    One wave per 16-col tile; ov_w1 (77 MB) streams at full 64B-segment
//    efficiency (each row segment touched by exactly one wave once).
// ---------------------------------------------------------------------------
__global__ void overall_gemm_wmma(const float* __restrict__ cat,  // [16,37632]
                                  const float* __restrict__ w,    // [37632,512]
                                  const float* __restrict__ bias, // [512]
                                  float* __restrict__ hidden2) {  // [16,512]
  const int lane  = threadIdx.x;
  const int n0    = blockIdx.x * 16;   // 32 N-tiles
  const int lrow  = lane & 15;
  const int khalf = (lane >> 4) << 1;
  const float* arow = cat + (size_t)lrow * CAT_K;
  v8f acc = {};
  for (int k = 0; k < CAT_K; k += 4) {
    const int ka = k + khalf;
    v2f a, bb;
    a.x  = arow[ka];
    a.y  = arow[ka + 1];
    bb.x = w[(size_t)ka * OVN + n0 + lrow];
    bb.y = w[(size_t)(ka + 1) * OVN + n0 + lrow];
    acc = __builtin_amdgcn_wmma_f32_16x16x4_f32(false, a, false, bb,
                                                (short)0, acc, false, false);
  }
  const int n  = n0 + lrow;
  const float bv = bias[n];
  const int mh = (lane >> 4) << 3;
#pragma unroll
  for (int j = 0; j < 8; ++j) {
    const int m = j + mh;
    const float v = acc[j] + bv;
    hidden2[(size_t)m * OVN + n] = v > 0.f ? v : 0.f;
  }
}

// ---------------------------------------------------------------------------
// 10) overall = hidden2 @ ov_w2 + ov_b2  (8 threads)
// ---------------------------------------------------------------------------
__global__ void overall_final(const float* __restrict__ hidden2,
                              const float* __restrict__ w2,  // [512]
                              const float* __restrict__ b2,  // [1]
                              float* __restrict__ out) {     // [8]
  const int im = threadIdx.x;
  if (im >= BIMG) return;
  const float* h = hidden2 + (size_t)im * OVN;
  float acc = b2[0];
  for (int j = 0; j < OVN; ++j) acc += h[j] * w2[j];
  out[im] = acc;
}

extern "C" void kernel_launch(void* const* d_in, const int* in_sizes, int n_in,
                              void* d_out, int out_size, void* d_ws, size_t ws_size,
                              hipStream_t stream) {
  const float* feat     = (const float*)d_in[0];
  const float* boxes[3] = {(const float*)d_in[1], (const float*)d_in[2],
                           (const float*)d_in[3]};
  const float* fpn_w  = (const float*)d_in[4];
  const float* fpn_b  = (const float*)d_in[5];
  const float* seg_w1 = (const float*)d_in[6];
  const float* seg_b1 = (const float*)d_in[7];
  const float* seg_w2 = (const float*)d_in[8];
  const float* seg_b2 = (const float*)d_in[9];
  const float* ov_w1  = (const float*)d_in[10];
  const float* ov_b1  = (const float*)d_in[11];
  const float* ov_w2  = (const float*)d_in[12];
  const float* ov_b2  = (const float*)d_in[13];

  // workspace layout (floats); total = 14,512,128 floats = ~58.0 MB
  float* ws   = (float*)d_ws;
  float* fpn  = ws;                    // 524288
  float* mid  = fpn  + 524288;         // 262144
  float* segs = mid  + 262144;         // 8192
  float* rf   = segs + 8192;           // 12845056  (reused across 3 heads)
  float* hid  = rf   + 12845056;       // 262144
  float* cat  = hid  + 262144;         // 16*37632 = 602112 (M padded to 16)
  float* hid2 = cat  + 602112;         // 8192

  float* out      = (float*)d_out;
  float* seg_out  = out;                       // 8*4*512*512 = 8388608
  float* qual_out = out + 8388608;             // 3*1024
  float* ov_out   = out + 8388608 + 3 * 1024;  // 8

  fpn_conv_wmma<<<dim3(16, 128), 32, 0, stream>>>(feat, fpn_w, fpn_b, fpn);
  seg_conv3x3<<<dim3(BIMG * CMID), 256, 0, stream>>>(fpn, seg_w1, seg_b1, mid);
  seg_conv1x1<<<dim3(32), 256, 0, stream>>>(mid, seg_w2, seg_b2, segs);
  seg_upsample<<<dim3(32768), 256, 0, stream>>>(segs, seg_out);

  for (int h = 0; h < 3; ++h) {
    const float* w1 = (const float*)d_in[14 + 4 * h];
    const float* b1 = (const float*)d_in[15 + 4 * h];
    const float* w2 = (const float*)d_in[16 + 4 * h];
    const float* b2 = (const float*)d_in[17 + 4 * h];
    roi_align_kernel<<<dim3(NBOXN), 256, 0, stream>>>(fpn, boxes[h], rf);
    qhead_gemm_wmma<<<dim3(4, 8), 256, 0, stream>>>(rf, w1, b1, hid);
    qual_kernel<<<dim3(4), 256, 0, stream>>>(hid, w2, b2, qual_out + h * NBOXN);
    seg_avg_kernel<<<dim3(49), 256, 0, stream>>>(rf, boxes[h], cat, h);
  }
  overall_gemm_wmma<<<dim3(32), 32, 0, stream>>>(cat, ov_w1, ov_b1, hid2);
  overall_final<<<1, 8, 0, stream>>>(hid2, ov_w2, ov_b2, ov_out);
}